// GNNModel_74002286510426
// MI455X (gfx1250) — compile-verified
//
#include <hip/hip_runtime.h>
#include <math.h>

// ---------------------------------------------------------------------------
// 3-layer GCN for MI455X (gfx1250, wave32).
//  - GEMMs via V_WMMA_F32_16X16X4_F32 (full fp32: GEMM is ~2.3 GFLOP total,
//    far from the bottleneck, so we buy accuracy instead of matrix TFLOPS).
//  - Edge aggregation via coalesced gathers + global_atomic_add_f32
//    (bandwidth/L2-atomic bound: ~3 GB worst case, features L2-resident).
// Workspace use: dinv[N] + 2 x N*64 f32 buffers  (~52 MB).
// ---------------------------------------------------------------------------

typedef __attribute__((ext_vector_type(2))) float v2f;
typedef __attribute__((ext_vector_type(8))) float v8f;

// C[M,Nout] = A[M,K] @ W[K,Nout], all row-major f32.
// Requires M%16==0, Nout%16==0, K%4==0 (true here: M=100000, N in {16,32,64},
// K in {256,32,64}). One wave per 16x16 C tile; W staged in LDS per block.
__global__ void __launch_bounds__(256) gcn_gemm_wmma_f32(
    const float* __restrict__ A, const float* __restrict__ W,
    float* __restrict__ C, int M, int K, int Nout) {
  extern __shared__ float lw[];  // K*Nout floats (max 32 KB for 256x32)
  const int tid = threadIdx.x;
  for (int i = tid; i < K * Nout; i += blockDim.x) lw[i] = W[i];
  __syncthreads();

  const int lane = tid & 31;
  const int wave = tid >> 5;
  const int wpb = blockDim.x >> 5;
  const int ntiles = Nout >> 4;
  const int gw = blockIdx.x * wpb + wave;
  const int mtile = gw / ntiles;
  if (mtile * 16 >= M) return;  // whole-wave exit: EXEC all-ones for WMMA
  const int n0 = (gw % ntiles) << 4;

  // ISA 32-bit A 16x4 layout: VGPR0/1 lanes 0-15 = K0,K1; lanes 16-31 = K2,K3.
  const int half = lane >> 4;
  const int l16 = lane & 15;
  const float* arow = A + (size_t)(mtile * 16 + l16) * K + half * 2;
  const float* wcol = lw + (size_t)(half * 2) * Nout + n0 + l16;

  v8f acc = {};
  for (int k0 = 0; k0 < K; k0 += 4) {
    v2f a = *(const v2f*)(arow + k0);       // 8B load: A[m][k0+2h], A[m][k0+2h+1]
    v2f b;
    b.x = wcol[(size_t)k0 * Nout];          // W[k0+2h][n]
    b.y = wcol[(size_t)(k0 + 1) * Nout];    // W[k0+2h+1][n]
    acc = __builtin_amdgcn_wmma_f32_16x16x4_f32(
        false, a, false, b, (short)0, acc, false, false);
  }
  // C/D layout: VGPR r -> lanes 0-15:(M=r,N=lane); lanes 16-31:(M=8+r,N=lane-16)
  float* crow = C + (size_t)(mtile * 16 + half * 8) * Nout + n0 + l16;
#pragma unroll
  for (int r = 0; r < 8; ++r) crow[(size_t)r * Nout] = acc[r];
}

__global__ void gcn_fill_f32(float* __restrict__ p, float v, long long n) {
  long long i = (long long)blockIdx.x * blockDim.x + threadIdx.x;
  if (i < n) p[i] = v;
}

__global__ void gcn_deg_accum(const long long* __restrict__ dst,
                              float* __restrict__ deg, long long E) {
  long long e = (long long)blockIdx.x * blockDim.x + threadIdx.x;
  if (e < E) atomicAdd(deg + (int)dst[e], 1.0f);
}

__global__ void gcn_deg_to_dinv(float* __restrict__ d, int n) {
  int i = blockIdx.x * blockDim.x + threadIdx.x;
  if (i < n) d[i] = rsqrtf(d[i]);
}

// out[dst*F + f] += h[src*F + f] * dinv[src]*dinv[dst]
// One thread per (edge, feature): F consecutive lanes share an edge, so the
// gather and the atomic both hit contiguous F*4-byte bursts (F in {16,32,64}).
__global__ void gcn_scatter_edges(const long long* __restrict__ src,
                                  const long long* __restrict__ dst,
                                  const float* __restrict__ h,
                                  const float* __restrict__ dinv,
                                  float* __restrict__ out, long long E,
                                  int logF) {
  long long gid = (long long)blockIdx.x * blockDim.x + threadIdx.x;
  if (gid >= (E << logF)) return;
  const int F = 1 << logF;
  long long e = gid >> logF;
  int f = (int)(gid & (F - 1));
  int s = (int)src[e];
  int d = (int)dst[e];
  float nrm = dinv[s] * dinv[d];
  atomicAdd(out + (((size_t)d) << logF) + f, h[(((size_t)s) << logF) + f] * nrm);
}

// out += h*dinv^2 + b ; optional ReLU
__global__ void gcn_finalize(const float* __restrict__ h,
                             const float* __restrict__ dinv,
                             const float* __restrict__ bias,
                             float* __restrict__ out, long long total, int logF,
                             int relu) {
  long long gid = (long long)blockIdx.x * blockDim.x + threadIdx.x;
  if (gid >= total) return;
  int i = (int)(gid >> logF);
  int f = (int)(gid & ((1 << logF) - 1));
  float di = dinv[i];
  float v = out[gid] + h[gid] * (di * di) + bias[f];
  out[gid] = relu ? fmaxf(v, 0.0f) : v;
}

__global__ void gcn_log_softmax(float* __restrict__ out, int n, int C) {
  int i = blockIdx.x * blockDim.x + threadIdx.x;
  if (i >= n) return;
  float v[16];
  float mx = -INFINITY;
  for (int c = 0; c < C; ++c) {
    v[c] = out[(size_t)i * C + c];
    mx = fmaxf(mx, v[c]);
  }
  float s = 0.f;
  for (int c = 0; c < C; ++c) s += expf(v[c] - mx);
  float lse = mx + logf(s);
  for (int c = 0; c < C; ++c) out[(size_t)i * C + c] = v[c] - lse;
}

static inline int gcn_blocks(long long n, int bs) {
  return (int)((n + bs - 1) / bs);
}
static inline int gcn_ilog2(int v) {
  int r = 0;
  while ((1 << (r + 1)) <= v) ++r;
  return r;
}

extern "C" void kernel_launch(void* const* d_in, const int* in_sizes, int n_in,
                              void* d_out, int out_size, void* d_ws,
                              size_t ws_size, hipStream_t stream) {
  const float* x = (const float*)d_in[0];
  const long long* ei = (const long long*)d_in[1];  // int64 [2, E]
  const float* W1 = (const float*)d_in[2];
  const float* b1 = (const float*)d_in[3];
  const float* W2 = (const float*)d_in[4];
  const float* b2 = (const float*)d_in[5];
  const float* W3 = (const float*)d_in[6];
  const float* b3 = (const float*)d_in[7];

  const int h1 = in_sizes[3];            // 32
  const int h2 = in_sizes[5];            // 64
  const int nc = in_sizes[7];            // 16
  const int fin = in_sizes[2] / h1;      // 256
  const int N = in_sizes[0] / fin;       // 100000
  const long long E = (long long)in_sizes[1] / 2;
  const long long* srcI = ei;
  const long long* dstI = ei + E;

  float* ws = (float*)d_ws;
  float* dinv = ws;                                   // N floats
  float* bufA = ws + (((size_t)N + 63) & ~(size_t)63);  // N*64 floats
  float* bufB = bufA + (size_t)N * 64;                  // N*64 floats
  float* outp = (float*)d_out;                          // N*nc floats

  const int l1 = gcn_ilog2(h1), l2 = gcn_ilog2(h2), l3 = gcn_ilog2(nc);
  const int mtiles = (N + 15) / 16;

  // --- degrees: deg = 1 + scatter(ones, dst); dinv = rsqrt(deg)
  gcn_fill_f32<<<gcn_blocks(N, 256), 256, 0, stream>>>(dinv, 1.0f, N);
  gcn_deg_accum<<<gcn_blocks(E, 256), 256, 0, stream>>>(dstI, dinv, E);
  gcn_deg_to_dinv<<<gcn_blocks(N, 256), 256, 0, stream>>>(dinv, N);

  // --- layer 1: H1 = X@W1 (bufA), O1 = agg(H1)+b1, ReLU (bufB)
  {
    long long waves = (long long)mtiles * (h1 / 16);
    gcn_gemm_wmma_f32<<<gcn_blocks(waves * 32, 256), 256,
                        (size_t)fin * h1 * 4, stream>>>(x, W1, bufA, N, fin, h1);
    long long n1 = (long long)N * h1;
    gcn_fill_f32<<<gcn_blocks(n1, 256), 256, 0, stream>>>(bufB, 0.f, n1);
    gcn_scatter_edges<<<gcn_blocks(E << l1, 256), 256, 0, stream>>>(
        srcI, dstI, bufA, dinv, bufB, E, l1);
    gcn_finalize<<<gcn_blocks(n1, 256), 256, 0, stream>>>(bufA, dinv, b1, bufB,
                                                          n1, l1, 1);
  }

  // --- layer 2: H2 = O1@W2 (bufA), O2 = agg(H2)+b2, ReLU (bufB)
  {
    long long waves = (long long)mtiles * (h2 / 16);
    gcn_gemm_wmma_f32<<<gcn_blocks(waves * 32, 256), 256,
                        (size_t)h1 * h2 * 4, stream>>>(bufB, W2, bufA, N, h1, h2);
    long long n2 = (long long)N * h2;
    gcn_fill_f32<<<gcn_blocks(n2, 256), 256, 0, stream>>>(bufB, 0.f, n2);
    gcn_scatter_edges<<<gcn_blocks(E << l2, 256), 256, 0, stream>>>(
        srcI, dstI, bufA, dinv, bufB, E, l2);
    gcn_finalize<<<gcn_blocks(n2, 256), 256, 0, stream>>>(bufA, dinv, b2, bufB,
                                                          n2, l2, 1);
  }

  // --- layer 3: H3 = O2@W3 (bufA), O3 = agg(H3)+b3 (d_out), log_softmax
  {
    long long waves = (long long)mtiles * (nc / 16);
    gcn_gemm_wmma_f32<<<gcn_blocks(waves * 32, 256), 256,
                        (size_t)h2 * nc * 4, stream>>>(bufB, W3, bufA, N, h2, nc);
    long long n3 = (long long)N * nc;
    gcn_fill_f32<<<gcn_blocks(n3, 256), 256, 0, stream>>>(outp, 0.f, n3);
    gcn_scatter_edges<<<gcn_blocks(E << l3, 256), 256, 0, stream>>>(
        srcI, dstI, bufA, dinv, outp, E, l3);
    gcn_finalize<<<gcn_blocks(n3, 256), 256, 0, stream>>>(bufA, dinv, b3, outp,
                                                          n3, l3, 0);
    gcn_log_softmax<<<gcn_blocks(N, 256), 256, 0, stream>>>(outp, N, nc);
  }
}